// EdgeAwareLoss_49692771615140
// MI455X (gfx1250) — compile-verified
//
#include <hip/hip_runtime.h>

// ---------------------------------------------------------------------------
// EdgeAwareLoss for MI455X (gfx1250, wave32).
// predictions: [32, 3, 512, 512] f32   targets: [32, 512, 512] int32
// out: scalar f32 = mean( CE(logsoftmax, tgt) * (1 + 2*(dilate^erode)) )
// Memory-bound (~134 MB -> ~5.8 us at 23.3 TB/s). Cross-lane reductions are
// done with V_WMMA_F32_16X16X4_F32 (A = partials, B = ones).
// ---------------------------------------------------------------------------

typedef float v2f __attribute__((ext_vector_type(2)));
typedef float v8f __attribute__((ext_vector_type(8)));

#define BB 32
#define CC 3
#define HH 512
#define WW 512
#define HWQ 65536            // (H*W)/4 quads per image
#define WQ  128              // quads per row
#define NPIX 8388608         // B*H*W
#define NQUAD 2097152        // NPIX/4

// Full 32-lane sum, result valid in every lane. Deterministic.
__device__ __forceinline__ float wave_sum32(float p) {
#if __has_builtin(__builtin_amdgcn_wmma_f32_16x16x4_f32)
    // A (16x4 f32, 2 VGPRs/lane): lane L<16 -> A[L][0],A[L][1]; lane L>=16 ->
    // A[L-16][2],A[L-16][3]. With a={p,0}: row m of A sums to p_m + p_{m+16}.
    // B = all ones (both VGPRs 1.0 in all lanes -> every element 1.0,
    // independent of the exact 4x16 layout). D[m][n] = p_m + p_{m+16}.
    v2f a; a.x = p; a.y = 0.0f;
    v2f b; b.x = 1.0f; b.y = 1.0f;
    v8f c = {};
    v8f d = __builtin_amdgcn_wmma_f32_16x16x4_f32(
        /*neg_a=*/false, a, /*neg_b=*/false, b,
        /*c_mod=*/(short)0, c, /*reuse_a=*/false, /*reuse_b=*/false);
    // D layout: lane<16: VGPR r = D[r][lane]; lane>=16: VGPR r = D[r+8][lane-16].
    float s = d[0] + d[1] + d[2] + d[3] + d[4] + d[5] + d[6] + d[7];
    // lanes 0-15: sum rows 0..7 ; lanes 16-31: sum rows 8..15 -> swap halves.
    s += __shfl_xor(s, 16, 32);
    return s;
#else
    for (int off = 16; off > 0; off >>= 1) p += __shfl_xor(p, off, 32);
    return p;
#endif
}

// Block (256 threads = 8 waves) sum; valid in thread 0. Requires all threads.
__device__ __forceinline__ float block_sum256(float p) {
    __shared__ float ws[8];
    float s = wave_sum32(p);
    const int lane = threadIdx.x & 31;
    const int wave = threadIdx.x >> 5;
    if (lane == 0) ws[wave] = s;
    __syncthreads();
    float r = 0.0f;
    if (threadIdx.x == 0) {
        #pragma unroll
        for (int i = 0; i < 8; ++i) r += ws[i];
    }
    __syncthreads();
    return r;
}

__global__ void __launch_bounds__(256)
edge_loss_partial(const float* __restrict__ pred,
                  const int* __restrict__ tgt,
                  float* __restrict__ partial) {
    const float4* __restrict__ predq = (const float4*)pred;
    const int4*   __restrict__ tgtq  = (const int4*)tgt;

    const int tid    = blockIdx.x * 256 + threadIdx.x;
    const int stride = gridDim.x * 256;

    float acc = 0.0f;

    for (int q = tid; q < NQUAD; q += stride) {
        const int b  = q >> 16;          // / HWQ
        const int rq = q & (HWQ - 1);
        const int h  = rq >> 7;          // / WQ
        const int wq = rq & (WQ - 1);

        const int4 t4 = tgtq[q];
        const bool hasUp   = (h > 0);
        const bool hasDown = (h < HH - 1);
        int4 u4 = make_int4(0, 0, 0, 0);
        int4 d4 = make_int4(0, 0, 0, 0);
        if (hasUp)   u4 = tgtq[q - WQ];
        if (hasDown) d4 = tgtq[q + WQ];
        const bool hasL = (wq > 0);
        const bool hasR = (wq < WQ - 1);
        int leftS = 0, rightS = 0;
        if (hasL) leftS  = tgt[4 * q - 1];
        if (hasR) rightS = tgt[4 * q + 4];

        const int baseq = b * (CC * HWQ) + rq;
        const float4 x0v = predq[baseq];
        const float4 x1v = predq[baseq + HWQ];
        const float4 x2v = predq[baseq + 2 * HWQ];

        const int   tA[4] = {t4.x, t4.y, t4.z, t4.w};
        const int   uA[4] = {u4.x, u4.y, u4.z, u4.w};
        const int   dA[4] = {d4.x, d4.y, d4.z, d4.w};
        const float x0A[4] = {x0v.x, x0v.y, x0v.z, x0v.w};
        const float x1A[4] = {x1v.x, x1v.y, x1v.z, x1v.w};
        const float x2A[4] = {x2v.x, x2v.y, x2v.z, x2v.w};

        #pragma unroll
        for (int j = 0; j < 4; ++j) {
            const int t = tA[j];
            int dmax = t, dmin = t;
            if (hasUp)   { const int v = uA[j]; dmax = max(dmax, v); dmin = min(dmin, v); }
            if (hasDown) { const int v = dA[j]; dmax = max(dmax, v); dmin = min(dmin, v); }
            // left neighbor
            if (j > 0)        { const int v = tA[j - 1]; dmax = max(dmax, v); dmin = min(dmin, v); }
            else if (hasL)    { const int v = leftS;     dmax = max(dmax, v); dmin = min(dmin, v); }
            // right neighbor
            if (j < 3)        { const int v = tA[j + 1]; dmax = max(dmax, v); dmin = min(dmin, v); }
            else if (hasR)    { const int v = rightS;    dmax = max(dmax, v); dmin = min(dmin, v); }

            const float wgt = 1.0f + 2.0f * (float)(dmax ^ dmin);

            const float x0 = x0A[j], x1 = x1A[j], x2 = x2A[j];
            const float m   = fmaxf(x0, fmaxf(x1, x2));
            const float sum = expf(x0 - m) + expf(x1 - m) + expf(x2 - m);
            const float lse = m + logf(sum);
            const float xt  = (t == 0) ? x0 : ((t == 1) ? x1 : x2);
            acc += (lse - xt) * wgt;
        }
    }

    const float bs = block_sum256(acc);
    if (threadIdx.x == 0) partial[blockIdx.x] = bs;
}

__global__ void __launch_bounds__(256)
edge_loss_final(const float* __restrict__ partial, int n, float* __restrict__ out) {
    float acc = 0.0f;
    for (int i = threadIdx.x; i < n; i += 256) acc += partial[i];
    const float s = block_sum256(acc);
    if (threadIdx.x == 0) out[0] = s * (1.0f / (float)NPIX);
}

extern "C" void kernel_launch(void* const* d_in, const int* in_sizes, int n_in,
                              void* d_out, int out_size, void* d_ws, size_t ws_size,
                              hipStream_t stream) {
    const float* pred = (const float*)d_in[0];
    const int*   tgt  = (const int*)d_in[1];
    float* out     = (float*)d_out;
    float* partial = (float*)d_ws;

    int nb = 1024;                                   // 262144 threads -> 8 quads each
    if (ws_size < (size_t)nb * sizeof(float)) nb = (int)(ws_size / sizeof(float));
    if (nb < 1) nb = 1;

    edge_loss_partial<<<nb, 256, 0, stream>>>(pred, tgt, partial);
    edge_loss_final<<<1, 256, 0, stream>>>(partial, nb, out);
}